// DigitCaps_5669356833323
// MI455X (gfx1250) — compile-verified
//
#include <hip/hip_runtime.h>
#include <math.h>

typedef __attribute__((ext_vector_type(2))) float v2f;
typedef __attribute__((ext_vector_type(8))) float v8f;

#define NB 512     // batch
#define NI 1152    // input capsules
#define NJ 10      // output capsules
#define DE 8       // D_IN
#define DD 16      // D_OUT
#define JD 160     // NJ*DD
#define VS_STRIDE 164  // LDS row stride (pad to spread banks)

// SWAPX16 swizzle: xor_mask=0x10, and_mask=0x1f -> offset 0x401F
__device__ __forceinline__ float xor16_add(float x) {
  int y = __builtin_amdgcn_ds_swizzle(__float_as_int(x), 0x401F);
  return x + __int_as_float(y);
}

// One wave per (16-batch-tile, i-chunk). Computes u_hat on the fly with
// V_WMMA_F32_16X16X4_F32 in transposed layout:
//   D[d' = r + 8*hi (+16*j), b = lo] = sum_e W[i,j,e,d'] * u[b0+b,i,e]
// then does routing softmax per lane and accumulates partial s.
__global__ __launch_bounds__(32) void caps_route_iter(
    const float* __restrict__ u,      // [512,1152,8]
    const float* __restrict__ W,      // [1152,10,8,16]
    const float* __restrict__ vsum,   // [512,160] cumulative v
    float* __restrict__ s_part,       // [nchunk,512,160]
    int ichunk, int iter)
{
  __shared__ float vs[16 * VS_STRIDE];
  const int lane = threadIdx.x;
  const int lo = lane & 15;
  const int hi = lane >> 4;
  const int b0 = blockIdx.x * 16;
  const int i0 = blockIdx.y * ichunk;

  if (iter != 0) {
    for (int k = lane; k < 16 * JD; k += 32) {
      int bb = k / JD, jd = k % JD;
      vs[bb * VS_STRIDE + jd] = vsum[(size_t)(b0 + bb) * JD + jd];
    }
  }
  __syncthreads();

  v8f s_acc[NJ];
#pragma unroll
  for (int j = 0; j < NJ; ++j) s_acc[j] = (v8f){};

  const float* ub = u + (size_t)(b0 + lo) * NI * DE;

  for (int ii = 0; ii < ichunk; ++ii) {
    const int i = i0 + ii;

    // B operand (u^T, 8x16): lane lo = batch col, e = v + 2*hi (+4*half)
    v2f bu0 = *(const v2f*)(ub + (size_t)i * DE + 2 * hi);
    v2f bu1 = *(const v2f*)(ub + (size_t)i * DE + 2 * hi + 4);

    if (ii + 1 < ichunk)  // warm L2/L0 for next capsule's weights
      __builtin_prefetch(W + (size_t)(i + 1) * NJ * DE * DD + lane * 40, 0, 1);

    v8f uh[NJ];
#pragma unroll
    for (int j = 0; j < NJ; ++j) {
      // A operand (W^T tile 16x8): lane lo = row d', e = v + 2*hi (+4*half)
      const float* wb = W + (((size_t)i * NJ + j) * DE + 2 * hi) * DD + lo;
      v2f a0; a0.x = wb[0];      a0.y = wb[DD];
      v2f a1; a1.x = wb[4 * DD]; a1.y = wb[5 * DD];
      v8f d = {};
      d = __builtin_amdgcn_wmma_f32_16x16x4_f32(false, a0, false, bu0,
                                                (short)0, d, false, false);
      d = __builtin_amdgcn_wmma_f32_16x16x4_f32(false, a1, false, bu1,
                                                (short)0, d, false, false);
      uh[j] = d;  // reg r, lane L: u_hat[b0+lo, i, j, d = r + 8*hi]
    }

    if (iter == 0) {
      // b == 0 -> softmax uniform == 0.1 exactly
#pragma unroll
      for (int j = 0; j < NJ; ++j)
#pragma unroll
        for (int r = 0; r < 8; ++r) s_acc[j][r] += 0.1f * uh[j][r];
    } else {
      float lg[NJ];
      const float* vrow = vs + lo * VS_STRIDE + 8 * hi;
#pragma unroll
      for (int j = 0; j < NJ; ++j) {
        float a = 0.f;
#pragma unroll
        for (int r = 0; r < 8; ++r) a = fmaf(uh[j][r], vrow[j * DD + r], a);
        lg[j] = xor16_add(a);  // combine d=0..7 (lanes<16) with d=8..15
      }
      float m = lg[0];
#pragma unroll
      for (int j = 1; j < NJ; ++j) m = fmaxf(m, lg[j]);
      float cc[NJ];
      float den = 0.f;
#pragma unroll
      for (int j = 0; j < NJ; ++j) { cc[j] = __expf(lg[j] - m); den += cc[j]; }
      float inv = __builtin_amdgcn_rcpf(den);
#pragma unroll
      for (int j = 0; j < NJ; ++j) {
        float cj = cc[j] * inv;
#pragma unroll
        for (int r = 0; r < 8; ++r) s_acc[j][r] = fmaf(cj, uh[j][r], s_acc[j][r]);
      }
    }
  }

  // partial s for this chunk: s_part[chunk][b][j][d]; each element owned by
  // exactly one (lane, reg) -> deterministic.
  float* sp = s_part + ((size_t)blockIdx.y * NB + b0 + lo) * JD + 8 * hi;
#pragma unroll
  for (int j = 0; j < NJ; ++j)
#pragma unroll
    for (int r = 0; r < 8; ++r) sp[j * DD + r] = s_acc[j][r];
}

// One block per batch row: reduce chunk partials, squash, update vsum.
__global__ __launch_bounds__(160) void caps_squash(
    const float* __restrict__ s_part,  // [nchunk,512,160]
    float* __restrict__ vsum,          // [512,160]
    float* __restrict__ out,           // [512,160]
    int nchunk, int iter, int last)
{
  __shared__ float sq[JD];
  const int b = blockIdx.x;
  const int t = threadIdx.x;        // t = j*16 + d
  float s = 0.f;
  for (int c = 0; c < nchunk; ++c)
    s += s_part[((size_t)c * NB + b) * JD + t];
  sq[t] = s * s;
  __syncthreads();
  const int j = t >> 4;
  float ss = 0.f;
#pragma unroll
  for (int d = 0; d < 16; ++d) ss += sq[j * 16 + d];
  float norm = sqrtf(ss);
  float factor = ss / (norm * (1.f + ss));   // matches reference squash
  float v = factor * s;
  float nv = (iter == 0) ? v : vsum[(size_t)b * JD + t] + v;
  vsum[(size_t)b * JD + t] = nv;
  if (last) out[(size_t)b * JD + t] = v;
}

extern "C" void kernel_launch(void* const* d_in, const int* in_sizes, int n_in,
                              void* d_out, int out_size, void* d_ws, size_t ws_size,
                              hipStream_t stream) {
  const float* u = (const float*)d_in[0];
  const float* W = (const float*)d_in[1];
  // d_in[2] is r == 3 (fixed in the reference)
  float* out = (float*)d_out;

  float* vsum = (float*)d_ws;                       // 512*160 floats
  float* s_part = vsum + (size_t)NB * JD;           // nchunk*512*160 floats

  int nchunk = 32;                                  // 1152/32 = 36 capsules/chunk
  while (nchunk > 1 &&
         (size_t)(nchunk + 1) * NB * JD * sizeof(float) > ws_size)
    nchunk >>= 1;
  const int ichunk = NI / nchunk;

  for (int t = 0; t < 3; ++t) {
    caps_route_iter<<<dim3(NB / 16, nchunk), 32, 0, stream>>>(
        u, W, vsum, s_part, ichunk, t);
    caps_squash<<<NB, JD, 0, stream>>>(s_part, vsum, out, nchunk, t, t == 2);
  }
}